// SAM_8117488189488
// MI455X (gfx1250) — compile-verified
//
#include <hip/hip_runtime.h>
#include <hip/hip_bf16.h>

// ---------------------------------------------------------------------------
// GNN (GCN x3 + attention pooling + MLP) for MI455X / gfx1250, wave32.
// f32 WMMA (V_WMMA_F32_16X16X4_F32): end-to-end time is bound by L2-resident
// edge scatter/atomics, so full-precision matrix math is free. W is staged
// TRANSPOSED in LDS so each B operand pair is one conflict-free ds_load_b64.
// ---------------------------------------------------------------------------

typedef __attribute__((ext_vector_type(2))) float v2f;
typedef __attribute__((ext_vector_type(8))) float v8f;

static constexpr int NN = 100000;   // nodes
static constexpr int EE = 1600000;  // edges
static constexpr int GG = 256;      // graphs

// ---------------------------------------------------------------------------
// WMMA GEMM: out[N,Fo] = A[N,K] @ W[K,Fo] (+bias)(+relu)
// blockDim = 256 (8 waves). Wave w owns rows [ (bx*8+w)*16 , +16 ).
// Each wave covers NT 16-wide column tiles starting at blockIdx.y*NT*16.
// REQUIRES N % 16 == 0 (true for all call sites: 100000, 256).
// ISA layouts (CDNA5 ISA §7.12.2, 32-bit):
//   A 16x4 : lanes 0-15 -> M=lane, VGPR r holds K = 2*half + r   (half=lane>>4)
//   B 4x16 : lanes 0-15 -> N=lane, VGPR r holds K = 2*half + r
//   C/D    : VGPR r, lanes 0-15 -> M=r, lanes 16-31 -> M=r+8; N=lane&15
// LDS holds W transposed: Wlds[c*KSTR + k], KSTR = K+4 (== 4 mod 8).
// B fetch = ds_load_b64 at (c*KSTR + ka)*4: across 32 lanes the 64 dwords
// land on all 64 banks exactly once (halves on disjoint mod-4 bank classes).
// ---------------------------------------------------------------------------
template <int NT, int K>
__global__ void gemm_wmma_f32(const float* __restrict__ A,
                              const float* __restrict__ W,
                              float* __restrict__ out,
                              const float* __restrict__ bias,
                              int N, int Fo, int do_relu) {
    constexpr int TC   = NT * 16;   // tile columns
    constexpr int KSTR = K + 4;     // padded per-column stride (dwords)
    __shared__ float Wlds[TC * KSTR];

    const int col0 = blockIdx.y * TC;

    // Cooperative transposed stage of W[0:K, col0:col0+TC] into LDS.
    // Global reads are coalesced (c fastest); LDS writes are strided.
    for (int idx = threadIdx.x; idx < K * TC; idx += 256) {
        const int k = idx / TC;
        const int c = idx - k * TC;
        Wlds[c * KSTR + k] = W[(size_t)k * Fo + col0 + c];
    }
    __syncthreads();

    const int lane = threadIdx.x & 31;
    const int wave = threadIdx.x >> 5;
    const int row0 = (blockIdx.x * 8 + wave) * 16;
    if (row0 >= N) return;          // wave-uniform, after the block barrier
    const int hh = lane >> 4;       // which half of the wave
    const int mn = lane & 15;       // M for A-load, N for B-load/D-store

    v8f acc[NT] = {};
    const float* Arow  = A + (size_t)(row0 + mn) * K;
    const float* Wlane = Wlds + mn * KSTR;   // column base for this lane

    for (int k0 = 0; k0 < K; k0 += 4) {
        const int ka = k0 + hh * 2;                 // even -> 8B aligned
        const v2f a = *(const v2f*)(Arow + ka);     // global_load_b64
#pragma unroll
        for (int t = 0; t < NT; ++t) {
            const v2f b = *(const v2f*)(Wlane + t * 16 * KSTR + ka); // ds_load_b64
            acc[t] = __builtin_amdgcn_wmma_f32_16x16x4_f32(
                false, a, false, b, (short)0, acc[t], false, false);
        }
    }

    const int rowD = row0 + hh * 8;
#pragma unroll
    for (int t = 0; t < NT; ++t) {
        const int c = col0 + t * 16 + mn;
        const float bv = bias ? bias[c] : 0.0f;
        float* op = out + (size_t)rowD * Fo + c;
#pragma unroll
        for (int r = 0; r < 8; ++r) {
            float v = acc[t][r] + bv;
            if (do_relu) v = fmaxf(v, 0.0f);
            op[(size_t)r * Fo] = v;
        }
    }
}

// ---------------------------------------------------------------------------
// Degree / normalization
// ---------------------------------------------------------------------------
__global__ void init_deg_ones(float* deg, int n) {
    int i = blockIdx.x * blockDim.x + threadIdx.x;
    if (i < n) deg[i] = 1.0f;   // self-loop contribution
}

__global__ void scatter_deg(const int* __restrict__ col, float* deg, int e) {
    int i = blockIdx.x * blockDim.x + threadIdx.x;
    if (i < e) atomicAdd(&deg[col[i]], 1.0f);
}

__global__ void compute_dinv(float* deg_dinv, int n) {
    int i = blockIdx.x * blockDim.x + threadIdx.x;
    if (i < n) {
        float d = deg_dinv[i];
        deg_dinv[i] = (d > 0.0f) ? rsqrtf(fmaxf(d, 1.0f)) : 0.0f;
    }
}

__global__ void compute_norm(const int* __restrict__ row,
                             const int* __restrict__ col,
                             const float* __restrict__ dinv,
                             float* __restrict__ normv, int e) {
    int i = blockIdx.x * blockDim.x + threadIdx.x;
    if (i < e) normv[i] = dinv[row[i]] * dinv[col[i]];
}

// ---------------------------------------------------------------------------
// Edge propagation: out[col] += h[row] * norm  (+ self loops with dinv^2).
// One wave per edge. Destinations pre-seeded with the layer bias; atomics
// resolve in the 192MB L2 (all feature buffers are <= 51MB).
// ---------------------------------------------------------------------------
__global__ void init_bias_bcast(float* __restrict__ out,
                                const float* __restrict__ bias,
                                size_t total, int Fmask) {
    size_t i = (size_t)blockIdx.x * blockDim.x + threadIdx.x;
    if (i < total) out[i] = bias[i & (size_t)Fmask];
}

// F == 128 fast path: one float4 gather per lane (global_load_b128).
__global__ void propagate128(const float* __restrict__ h,
                             const int* __restrict__ row,
                             const int* __restrict__ col,
                             const float* __restrict__ normv,
                             const float* __restrict__ dinv,
                             float* __restrict__ out, int e, int n) {
    const long long wid = ((long long)blockIdx.x * blockDim.x + threadIdx.x) >> 5;
    const int lane = threadIdx.x & 31;
    if (wid >= (long long)e + n) return;
    int r, c; float nv;
    if (wid < e) {
        r = row[wid]; c = col[wid]; nv = normv[wid];
    } else {
        r = c = (int)(wid - e);
        float d = dinv[r]; nv = d * d;
    }
    const float4 hv = ((const float4*)(h + (size_t)r * 128))[lane];
    float* oc = out + (size_t)c * 128 + lane * 4;
    atomicAdd(oc + 0, hv.x * nv);
    atomicAdd(oc + 1, hv.y * nv);
    atomicAdd(oc + 2, hv.z * nv);
    atomicAdd(oc + 3, hv.w * nv);
}

// generic (F = 64 / 32)
__global__ void propagate(const float* __restrict__ h,
                          const int* __restrict__ row,
                          const int* __restrict__ col,
                          const float* __restrict__ normv,
                          const float* __restrict__ dinv,
                          float* __restrict__ out, int e, int n, int F) {
    const long long wid = ((long long)blockIdx.x * blockDim.x + threadIdx.x) >> 5;
    const int lane = threadIdx.x & 31;
    if (wid >= (long long)e + n) return;
    int r, c; float nv;
    if (wid < e) {
        r = row[wid]; c = col[wid]; nv = normv[wid];
    } else {
        r = c = (int)(wid - e);
        float d = dinv[r]; nv = d * d;
    }
    const float* hr = h + (size_t)r * F;
    float* oc = out + (size_t)c * F;
    for (int f = lane; f < F; f += 32) atomicAdd(&oc[f], hr[f] * nv);
}

__global__ void propagate_f1(const float* __restrict__ h,
                             const int* __restrict__ row,
                             const int* __restrict__ col,
                             const float* __restrict__ normv,
                             const float* __restrict__ dinv,
                             float* __restrict__ out, int e, int n) {
    const long long i = (long long)blockIdx.x * blockDim.x + threadIdx.x;
    if (i >= (long long)e + n) return;
    int r, c; float nv;
    if (i < e) { r = row[i]; c = col[i]; nv = normv[i]; }
    else       { r = c = (int)(i - e); float d = dinv[r]; nv = d * d; }
    atomicAdd(&out[c], h[r] * nv);
}

__global__ void relu_k(float* x, size_t n) {
    size_t i = (size_t)blockIdx.x * blockDim.x + threadIdx.x;
    if (i < n) x[i] = fmaxf(x[i], 0.0f);
}

__global__ void zero_k(float* p, size_t n) {
    size_t i = (size_t)blockIdx.x * blockDim.x + threadIdx.x;
    if (i < n) p[i] = 0.0f;
}

// ---------------------------------------------------------------------------
// Attention head 32->1 (tiny GEMV)
// ---------------------------------------------------------------------------
__global__ void att_dot32(const float* __restrict__ a,
                          const float* __restrict__ w,
                          float* __restrict__ out, int n) {
    int i = blockIdx.x * blockDim.x + threadIdx.x;
    if (i < n) {
        const float* ar = a + (size_t)i * 32;
        float s = 0.0f;
#pragma unroll
        for (int k = 0; k < 32; ++k) s += ar[k] * w[k];
        out[i] = s;
    }
}

// ---------------------------------------------------------------------------
// Segment softmax + attention pooling. Weights are post-ReLU (>=0) and m is
// seeded with 0, so integer atomicMax on raw float bits is order-preserving
// and matches the reference's finite-guard (empty segments -> 0).
// ---------------------------------------------------------------------------
__global__ void seg_max_k(const float* __restrict__ w,
                          const int* __restrict__ batch,
                          float* m, int n) {
    int i = blockIdx.x * blockDim.x + threadIdx.x;
    if (i < n) atomicMax((int*)&m[batch[i]], __float_as_int(w[i]));
}

__global__ void seg_expsum_k(const float* __restrict__ w,
                             const int* __restrict__ batch,
                             const float* __restrict__ m,
                             float* __restrict__ e, float* s, int n) {
    int i = blockIdx.x * blockDim.x + threadIdx.x;
    if (i < n) {
        int g = batch[i];
        float v = expf(w[i] - m[g]);
        e[i] = v;
        atomicAdd(&s[g], v);
    }
}

__global__ void pool_k(const float* __restrict__ x,
                       const float* __restrict__ e,
                       const float* __restrict__ s,
                       const int* __restrict__ batch,
                       float* __restrict__ pooled, int n) {
    const long long wid = ((long long)blockIdx.x * blockDim.x + threadIdx.x) >> 5;
    const int lane = threadIdx.x & 31;
    if (wid >= n) return;
    const int g = batch[wid];
    const float wn = e[wid] / (s[g] + 1e-16f);
    const float4 xv = ((const float4*)(x + (size_t)wid * 128))[lane];
    float* pg = pooled + (size_t)g * 128 + lane * 4;
    atomicAdd(pg + 0, xv.x * wn);
    atomicAdd(pg + 1, xv.y * wn);
    atomicAdd(pg + 2, xv.z * wn);
    atomicAdd(pg + 3, xv.w * wn);
}

// ---------------------------------------------------------------------------
// Host-side orchestration
// ---------------------------------------------------------------------------
static inline unsigned int nblk(size_t n, int b) { return (unsigned int)((n + b - 1) / b); }

extern "C" void kernel_launch(void* const* d_in, const int* in_sizes, int n_in,
                              void* d_out, int out_size, void* d_ws, size_t ws_size,
                              hipStream_t stream) {
    (void)in_sizes; (void)n_in; (void)out_size; (void)ws_size;

    const float* x     = (const float*)d_in[0];
    const int*   eidx  = (const int*)d_in[1];
    const int*   batch = (const int*)d_in[2];
    const float* W1 = (const float*)d_in[3];  const float* b1 = (const float*)d_in[4];
    const float* W2 = (const float*)d_in[5];  const float* b2 = (const float*)d_in[6];
    const float* W3 = (const float*)d_in[7];  const float* b3 = (const float*)d_in[8];
    const float* Wa1 = (const float*)d_in[9];  const float* ba1 = (const float*)d_in[10];
    const float* Wa2 = (const float*)d_in[11]; const float* ba2 = (const float*)d_in[12];
    const float* Wa3 = (const float*)d_in[13]; const float* ba3 = (const float*)d_in[14];
    const float* Wm1 = (const float*)d_in[15]; const float* bm1 = (const float*)d_in[16];
    const float* Wm2 = (const float*)d_in[17]; const float* bm2 = (const float*)d_in[18];
    float* out = (float*)d_out;

    const int* row = eidx;          // edge_index[0]
    const int* col = eidx + EE;     // edge_index[1]

    // workspace carve-up (floats)
    float* ws = (float*)d_ws;
    size_t o = 0;
    float* buf_h  = ws + o; o += (size_t)NN * 128;   // GEMM scratch
    float* buf_x0 = ws + o; o += (size_t)NN * 128;   // ping
    float* buf_x1 = ws + o; o += (size_t)NN * 128;   // pong / attention chain
    float* dinv   = ws + o; o += NN;                 // deg -> dinv in place
    float* normv  = ws + o; o += EE;
    float* mbuf   = ws + o; o += GG;
    float* sbuf   = ws + o; o += GG;
    float* pooled = ws + o; o += (size_t)GG * 128;
    float* mlph   = ws + o; o += (size_t)GG * 128;

    const int rowBlocksN = (NN + 127) / 128;   // 8 waves x 16 rows per block
    const long long edgeThreads = ((long long)EE + NN) * 32;

    // --- gcn_norm: deg -> dinv -> per-edge norm -----------------------------
    init_deg_ones<<<nblk(NN, 256), 256, 0, stream>>>(dinv, NN);
    scatter_deg<<<nblk(EE, 256), 256, 0, stream>>>(col, dinv, EE);
    compute_dinv<<<nblk(NN, 256), 256, 0, stream>>>(dinv, NN);
    compute_norm<<<nblk(EE, 256), 256, 0, stream>>>(row, col, dinv, normv, EE);

    // --- shared post-GEMM steps of one GCN layer ---------------------------
    auto post = [&](const float* hbuf, float* obuf, const float* b, int Fo) {
        size_t tot = (size_t)NN * Fo;
        init_bias_bcast<<<nblk(tot, 256), 256, 0, stream>>>(obuf, b, tot, Fo - 1);
        if (Fo == 128)
            propagate128<<<nblk((size_t)edgeThreads, 256), 256, 0, stream>>>(hbuf, row, col, normv, dinv, obuf, EE, NN);
        else
            propagate<<<nblk((size_t)edgeThreads, 256), 256, 0, stream>>>(hbuf, row, col, normv, dinv, obuf, EE, NN, Fo);
        relu_k<<<nblk(tot, 256), 256, 0, stream>>>(obuf, tot);
    };

    // --- 3x GCN(128->128) ---------------------------------------------------
    gemm_wmma_f32<8, 128><<<dim3(rowBlocksN, 1), 256, 0, stream>>>(x, W1, buf_h, nullptr, NN, 128, 0);
    post(buf_h, buf_x0, b1, 128);
    gemm_wmma_f32<8, 128><<<dim3(rowBlocksN, 1), 256, 0, stream>>>(buf_x0, W2, buf_h, nullptr, NN, 128, 0);
    post(buf_h, buf_x1, b2, 128);
    gemm_wmma_f32<8, 128><<<dim3(rowBlocksN, 1), 256, 0, stream>>>(buf_x1, W3, buf_h, nullptr, NN, 128, 0);
    post(buf_h, buf_x0, b3, 128);                    // x_final = buf_x0

    // --- attention chain 128->64->32->1 ------------------------------------
    float* a1 = buf_x1;                      // N x 64
    float* a2 = buf_x1 + (size_t)NN * 64;    // N x 32
    float* wn = buf_x1 + (size_t)NN * 96;    // N x 1
    gemm_wmma_f32<4, 128><<<dim3(rowBlocksN, 1), 256, 0, stream>>>(buf_x0, Wa1, buf_h, nullptr, NN, 64, 0);
    post(buf_h, a1, ba1, 64);
    gemm_wmma_f32<2, 64><<<dim3(rowBlocksN, 1), 256, 0, stream>>>(a1, Wa2, buf_h, nullptr, NN, 32, 0);
    post(buf_h, a2, ba2, 32);

    att_dot32<<<nblk(NN, 256), 256, 0, stream>>>(a2, Wa3, buf_h, NN);
    init_bias_bcast<<<nblk(NN, 256), 256, 0, stream>>>(wn, ba3, NN, 0);
    propagate_f1<<<nblk((size_t)EE + NN, 256), 256, 0, stream>>>(buf_h, row, col, normv, dinv, wn, EE, NN);
    relu_k<<<nblk(NN, 256), 256, 0, stream>>>(wn, NN);

    // --- segment softmax + pooling -----------------------------------------
    zero_k<<<1, 256, 0, stream>>>(mbuf, GG);
    zero_k<<<1, 256, 0, stream>>>(sbuf, GG);
    zero_k<<<nblk((size_t)GG * 128, 256), 256, 0, stream>>>(pooled, (size_t)GG * 128);
    seg_max_k<<<nblk(NN, 256), 256, 0, stream>>>(wn, batch, mbuf, NN);
    seg_expsum_k<<<nblk(NN, 256), 256, 0, stream>>>(wn, batch, mbuf, buf_h, sbuf, NN);
    pool_k<<<nblk((size_t)NN * 32, 256), 256, 0, stream>>>(buf_x0, buf_h, sbuf, batch, pooled, NN);

    // --- MLP head: relu(pooled@Wm1+bm1) @ Wm2 + bm2 -> out [256,256] -------
    const int rowBlocksG = (GG + 127) / 128;
    gemm_wmma_f32<8, 128><<<dim3(rowBlocksG, 1), 256, 0, stream>>>(pooled, Wm1, mlph, bm1, GG, 128, 1);
    gemm_wmma_f32<8, 128><<<dim3(rowBlocksG, 2), 256, 0, stream>>>(mlph,   Wm2, out,  bm2, GG, 256, 0);
}